// BaseConvPlus_206158430693
// MI455X (gfx1250) — compile-verified
//
#include <hip/hip_runtime.h>

// Problem constants (from reference)
#define BB   32
#define CI   32
#define CO   32
#define KK   3
#define HH   192
#define WW   192
#define HW   (HH * WW)
#define TS   32          // spatial tile
#define STRD 36          // LDS tile row stride (16B aligned, padded)

typedef __attribute__((ext_vector_type(2))) float v2f;
typedef __attribute__((ext_vector_type(4))) float v4f;
typedef __attribute__((ext_vector_type(8))) float v8f;

typedef __attribute__((address_space(1))) int gint_t;   // global
typedef __attribute__((address_space(3))) int lint_t;   // LDS

#if __has_builtin(__builtin_amdgcn_global_load_async_to_lds_b32) && \
    __has_builtin(__builtin_amdgcn_s_wait_asynccnt)
#define ASYNC_LDS 1
#else
#define ASYNC_LDS 0
#endif

// ---------------------------------------------------------------------------
// Stage 1: kernel seed.  mean(depthwise_conv(x, wk)) collapses to border sums.
//   T[p][q] = total - excludedRow - excludedCol + corner
//   kernel[b,c,j] = (1/HW) * sum_{p,q} wk[(c*9+j),p,q] * T[p][q]
// ---------------------------------------------------------------------------
__global__ __launch_bounds__(256) void k_seed(const float* __restrict__ x,
                                              const float* __restrict__ wk,
                                              float* __restrict__ ker) {
  const int blk = blockIdx.x;                 // b*CI + c
  const float* img = x + (size_t)blk * HW;

  float s0 = 0.f, s1 = 0.f, s2 = 0.f, s3 = 0.f, s4 = 0.f; // tot,row0,rowL,col0,colL
  for (int idx = threadIdx.x; idx < HW; idx += 256) {
    float v = img[idx];
    int r  = idx / WW;
    int cc = idx - r * WW;
    s0 += v;
    if (r  == 0)      s1 += v;
    if (r  == HH - 1) s2 += v;
    if (cc == 0)      s3 += v;
    if (cc == WW - 1) s4 += v;
  }
  // wave32 reduction
  #pragma unroll
  for (int off = 16; off > 0; off >>= 1) {
    s0 += __shfl_down(s0, off);
    s1 += __shfl_down(s1, off);
    s2 += __shfl_down(s2, off);
    s3 += __shfl_down(s3, off);
    s4 += __shfl_down(s4, off);
  }
  __shared__ float part[5][8];
  const int wid = threadIdx.x >> 5, lane = threadIdx.x & 31;
  if (lane == 0) {
    part[0][wid] = s0; part[1][wid] = s1; part[2][wid] = s2;
    part[3][wid] = s3; part[4][wid] = s4;
  }
  __syncthreads();

  __shared__ float Ts[9];
  if (threadIdx.x == 0) {
    float tot = 0.f, r0 = 0.f, rL = 0.f, c0 = 0.f, cL = 0.f;
    #pragma unroll
    for (int w8 = 0; w8 < 8; ++w8) {
      tot += part[0][w8]; r0 += part[1][w8]; rL += part[2][w8];
      c0  += part[3][w8]; cL += part[4][w8];
    }
    const float x00 = img[0];
    const float x0L = img[WW - 1];
    const float xL0 = img[(size_t)(HH - 1) * WW];
    const float xLL = img[(size_t)HW - 1];
    #pragma unroll
    for (int p = 0; p < 3; ++p) {
      #pragma unroll
      for (int q = 0; q < 3; ++q) {
        float rEx = (p == 0) ? rL : ((p == 2) ? r0 : 0.f);
        float cEx = (q == 0) ? cL : ((q == 2) ? c0 : 0.f);
        float corner = 0.f;
        if (p == 0 && q == 0) corner = xLL;
        if (p == 0 && q == 2) corner = xL0;
        if (p == 2 && q == 0) corner = x0L;
        if (p == 2 && q == 2) corner = x00;
        Ts[p * 3 + q] = tot - rEx - cEx + corner;
      }
    }
  }
  __syncthreads();

  if (threadIdx.x < 9) {
    const int c = blk & (CI - 1);
    const float* wr = wk + (size_t)(c * 9 + threadIdx.x) * 9;
    float acc = 0.f;
    #pragma unroll
    for (int t = 0; t < 9; ++t) acc += wr[t] * Ts[t];
    ker[(size_t)blk * 9 + threadIdx.x] = acc * (1.0f / (float)HW);
  }
}

// ---------------------------------------------------------------------------
// Stage 2a: win[b,o,j] = sum_i ker[b,i,j]*w_in[o,i] + b_in[o]
// ---------------------------------------------------------------------------
__global__ __launch_bounds__(288) void k_win(const float* __restrict__ ker,
                                             const float* __restrict__ w_in,
                                             const float* __restrict__ b_in,
                                             float* __restrict__ win) {
  const int b = blockIdx.x;
  __shared__ float sk[288];
  const int t = threadIdx.x;
  sk[t] = ker[(size_t)b * 288 + t];
  __syncthreads();
  const int o = t / 9, j = t - o * 9;
  float acc = b_in[o];
  #pragma unroll 8
  for (int i = 0; i < CI; ++i) acc += sk[i * 9 + j] * w_in[o * CI + i];
  win[(size_t)b * 288 + t] = acc;
}

// ---------------------------------------------------------------------------
// Stage 2b: wout[32x32] = kerF[32x288] * w_outF^T[288x32]  via f32 WMMA.
// One block, 4 waves; wave (mi,ni) owns one 16x16 tile; 72 K-steps of 4.
// A 16x4 f32 layout: lanes0-15 VGPR{0,1}=K{0,1}; lanes16-31 VGPR{0,1}=K{2,3}.
// ---------------------------------------------------------------------------
__global__ __launch_bounds__(128) void k_wout(const float* __restrict__ ker,
                                              const float* __restrict__ w_out,
                                              float* __restrict__ wout) {
  const int lane = threadIdx.x & 31;
  const int wid  = threadIdx.x >> 5;
  const int mi = (wid >> 1) & 1;
  const int ni = wid & 1;
  const int lm = lane & 15;
  const int hi = lane >> 4;                 // 0: K pair {0,1}, 1: {2,3}
  const int row = 16 * mi + lm;             // A row  (batch index)
  const int col = 16 * ni + lm;             // B col  (output channel)

  v8f c = {};
  for (int k0 = 0; k0 < 288; k0 += 4) {
    const int ka = k0 + 2 * hi;
    v2f a = { ker[(size_t)row * 288 + ka],   ker[(size_t)row * 288 + ka + 1] };
    v2f b = { w_out[(size_t)col * 288 + ka], w_out[(size_t)col * 288 + ka + 1] };
    // D = A x B + C   (f32, 16x16x4)
    c = __builtin_amdgcn_wmma_f32_16x16x4_f32(
        /*neg_a=*/false, a, /*neg_b=*/false, b,
        /*c_mod=*/(short)0, c, /*reuse_a=*/false, /*reuse_b=*/false);
  }
  #pragma unroll
  for (int v = 0; v < 8; ++v) {
    const int m = 16 * mi + v + 8 * hi;     // C/D layout: VGPR v -> M=v / v+8
    const int n = 16 * ni + lm;
    wout[m * CO + n] = c[v];
  }
}

// ---------------------------------------------------------------------------
// Stage 3: Z[b] = conv(x[b], win[b]) over CIN channels (SAME),
//          out[b,o] = wout[b,o] * Z[b].    Memory-bound: read x, write out.
//
// Double-buffered per-channel 34x34 halo tiles in LDS, filled with CDNA5
// GLOBAL_LOAD_ASYNC_TO_LDS (ASYNCcnt) when available.  OOB halo slots are the
// same lanes for every channel, so both buffers are zeroed once at entry and
// OOB lanes simply skip the async load (EXEC-masked) -> zeros persist.
// Output uses non-temporal stores so the 151 MB write-once stream does not
// evict x from the 192 MB L2 (x is read twice across k_seed/k_conv).
// ---------------------------------------------------------------------------
__global__ __launch_bounds__(256) void k_conv(const float* __restrict__ x,
                                              const float* __restrict__ win,
                                              const float* __restrict__ wout,
                                              float* __restrict__ out) {
  const int b  = blockIdx.y;
  const int ty = blockIdx.x / (WW / TS);
  const int tx = blockIdx.x - ty * (WW / TS);
  const int y0 = ty * TS, x0 = tx * TS;

  __shared__ float sw[288];
  __shared__ float swo[CO];
  __shared__ __align__(16) float tileS[2][34 * STRD];

  const int tid = threadIdx.x;
  if (tid < 288) sw[tid] = win[(size_t)b * 288 + tid];
  if (tid < CO)  swo[tid] = wout[b * CO + tid];

  // zero both buffers once (covers halo slots of border tiles)
  for (int idx = tid; idx < 2 * 34 * STRD; idx += 256)
    (&tileS[0][0])[idx] = 0.f;

  const float* xb = x + (size_t)b * CI * HW;
  const int r  = tid >> 3;          // 0..31 tile row
  const int cb = (tid & 7) << 2;    // 0,4,...,28 tile col base (4 px/thread)

  auto issue_tile = [&](int i, int buf) {
    const float* ch = xb + (size_t)i * HW;
    for (int idx = tid; idx < 34 * 34; idx += 256) {
      const int ly = idx / 34, lx = idx - ly * 34;
      const int gy = y0 - 1 + ly, gx = x0 - 1 + lx;
      if (gy >= 0 && gy < HH && gx >= 0 && gx < WW) {
#if ASYNC_LDS
        __builtin_amdgcn_global_load_async_to_lds_b32(
            (gint_t*)(ch + (size_t)gy * WW + gx),
            (lint_t*)&tileS[buf][ly * STRD + lx],
            /*offset=*/0, /*cpol=*/0);
#else
        tileS[buf][ly * STRD + lx] = ch[(size_t)gy * WW + gx];
#endif
      }
    }
  };

  __syncthreads();                 // zeros visible before first async fill
  issue_tile(0, 0);
#if ASYNC_LDS
  __builtin_amdgcn_s_wait_asynccnt(0);
#endif
  __syncthreads();

  float z0 = 0.f, z1 = 0.f, z2 = 0.f, z3 = 0.f;

  for (int i = 0; i < CI; ++i) {
    const int cur = i & 1;
    if (i + 1 < CI) issue_tile(i + 1, cur ^ 1);   // overlap next fill w/ compute

    const float* wrow = &sw[i * 9];
    #pragma unroll
    for (int p = 0; p < 3; ++p) {
      const float* lrow = &tileS[cur][(r + p) * STRD + cb];
      const float4 v4 = *(const float4*)lrow;          // px 0..3
      const float2 v2 = *(const float2*)(lrow + 4);    // px 4..5
      const float w0 = wrow[p * 3 + 0];
      const float w1 = wrow[p * 3 + 1];
      const float w2 = wrow[p * 3 + 2];
      z0 += w0 * v4.x + w1 * v4.y + w2 * v4.z;
      z1 += w0 * v4.y + w1 * v4.z + w2 * v4.w;
      z2 += w0 * v4.z + w1 * v4.w + w2 * v2.x;
      z3 += w0 * v4.w + w1 * v2.x + w2 * v2.y;
    }

#if ASYNC_LDS
    __builtin_amdgcn_s_wait_asynccnt(0);
#endif
    __syncthreads();
  }

  // rank-1 expansion over output channels; NT float4 stores (write-once data)
  float* ob = out + (size_t)b * CO * HW + (size_t)(y0 + r) * WW + (x0 + cb);
  #pragma unroll 4
  for (int o = 0; o < CO; ++o) {
    const float s = swo[o];
    v4f v = { s * z0, s * z1, s * z2, s * z3 };
    __builtin_nontemporal_store(v, (v4f*)(ob + (size_t)o * HW));
  }
}

// ---------------------------------------------------------------------------
extern "C" void kernel_launch(void* const* d_in, const int* in_sizes, int n_in,
                              void* d_out, int out_size, void* d_ws, size_t ws_size,
                              hipStream_t stream) {
  (void)in_sizes; (void)n_in; (void)out_size; (void)ws_size;
  const float* x     = (const float*)d_in[0];
  const float* wk    = (const float*)d_in[1];
  const float* w_in  = (const float*)d_in[2];
  const float* b_in  = (const float*)d_in[3];
  const float* w_out = (const float*)d_in[4];
  float* out = (float*)d_out;

  float* ker  = (float*)d_ws;          // [B*CI*9]  = 9216 floats
  float* winw = ker  + BB * CI * 9;    // [B*CI*9]  = 9216 floats
  float* wo   = winw + BB * CI * 9;    // [B*CO]    = 1024 floats

  k_seed<<<BB * CI, 256, 0, stream>>>(x, wk, ker);
  k_win <<<BB, 288, 0, stream>>>(ker, w_in, b_in, winw);
  k_wout<<<1, 128, 0, stream>>>(ker, w_out, wo);
  dim3 grid((HH / TS) * (WW / TS), BB);
  k_conv<<<grid, 256, 0, stream>>>(x, winw, wo, out);
}